// ImageAttentionResnetModule_50929722196574
// MI455X (gfx1250) — compile-verified
//
#include <hip/hip_runtime.h>

typedef float v2f __attribute__((ext_vector_type(2)));
typedef float v8f __attribute__((ext_vector_type(8)));

#define IN_EPS 1e-5f

// ---------------------------------------------------------------------------
// Per-(b,c) instance-norm statistics: mean and rsqrt(var+eps).
// One workgroup per (b,c) plane; grid-stride + LDS tree reduction.
// ---------------------------------------------------------------------------
__global__ __launch_bounds__(256) void stats_kernel(
    const float* __restrict__ x, float* __restrict__ mean,
    float* __restrict__ inv, int HW)
{
    int bc = blockIdx.x;
    const float* xp = x + (size_t)bc * HW;
    float s = 0.0f, s2 = 0.0f;
    for (int i = threadIdx.x; i < HW; i += 256) {
        float v = xp[i];
        s += v;
        s2 = fmaf(v, v, s2);
    }
    __shared__ float r1[256];
    __shared__ float r2[256];
    int tid = threadIdx.x;
    r1[tid] = s;
    r2[tid] = s2;
    __syncthreads();
    for (int o = 128; o > 0; o >>= 1) {
        if (tid < o) { r1[tid] += r1[tid + o]; r2[tid] += r2[tid + o]; }
        __syncthreads();
    }
    if (tid == 0) {
        float m = r1[0] / (float)HW;
        float v = r2[0] / (float)HW - m * m;   // biased variance (ddof=0)
        mean[bc] = m;
        inv[bc]  = rsqrtf(v + IN_EPS);
    }
}

// ---------------------------------------------------------------------------
// Fused A-element fetch for the implicit GEMM: gather + zero-pad +
// (optional) instance-norm + (optional) ReLU applied on the fly.
// ---------------------------------------------------------------------------
template<bool NORM, bool RELU>
__device__ __forceinline__ float load_a(
    const float* __restrict__ in, const float* __restrict__ nmean,
    const float* __restrict__ ninv, int cb, int statbase,
    int iy0, int ix0, int c, int ky, int kx,
    int Cin, int Hi, int Wi, int HiWi)
{
    int iy = iy0 + ky;
    int ix = ix0 + kx;
    float v = 0.0f;
    if ((c < Cin) & ((unsigned)iy < (unsigned)Hi) & ((unsigned)ix < (unsigned)Wi)) {
        v = in[cb + c * HiWi + iy * Wi + ix];
        if (NORM) {
            v = (v - nmean[statbase + c]) * ninv[statbase + c];
            if (RELU) v = fmaxf(v, 0.0f);
        }
    }
    return v;
}

// ---------------------------------------------------------------------------
// Implicit-GEMM convolution on the WMMA pipe (f32 16x16x4, full precision).
//   M = flattened (b, oy, ox) output positions, N = 32 output channels,
//   K = Cin*KH*KW (zero-padded to a multiple of 4).
// One wave computes a 64(M) x 32(N) tile: 4 M-frags x 2 N-frags per K-step.
// Weights (transposed to [k][n]) and the k->(c,ky,kx) decode table live in LDS.
// Bias add and optional in-place residual accumulation are fused into the
// D-matrix store. Requires: M_total % 512 == 0 and (Ho*Wo) % 64 == 0 (true
// for every layer of this network), so all waves are fully valid and a wave
// never straddles a batch boundary.
// ---------------------------------------------------------------------------
template<bool NORM, bool RELU, bool RESIN>
__global__ __launch_bounds__(256) void conv_wmma_kernel(
    const float* __restrict__ in, const float* __restrict__ w,
    const float* __restrict__ bias,
    const float* __restrict__ nmean, const float* __restrict__ ninv,
    float* __restrict__ out,
    int Cin, int KH, int KW, int stride, int pad,
    int Hi, int Wi, int Ho, int Wo, int Ksz, int Kpad)
{
    __shared__ float ldsW[288 * 32];   // weights as W[k][n], zero-padded
    __shared__ int   ldsT[288];        // packed (c<<16)|(ky<<8)|kx per k

    int tid = threadIdx.x;
    int KHW = KH * KW;
    for (int idx = tid; idx < (Kpad << 5); idx += 256) {
        int k = idx >> 5, n = idx & 31;
        ldsW[idx] = (k < Ksz) ? w[n * Ksz + k] : 0.0f;
    }
    for (int k = tid; k < Kpad; k += 256) {
        int packed;
        if (k < Ksz) {
            int c  = k / KHW;
            int rr = k - c * KHW;
            int ky = rr / KW;
            int kx = rr - ky * KW;
            packed = (c << 16) | (ky << 8) | kx;
        } else {
            packed = 0x40000000;       // c = 0x4000 -> always out of range
        }
        ldsT[k] = packed;
    }
    __syncthreads();

    int lane = tid & 31;
    int wave = tid >> 5;
    int lo   = lane & 15;
    int hi   = lane >> 4;
    int HoWo = Ho * Wo;
    int HiWi = Hi * Wi;
    int Mbase = (blockIdx.x * 8 + wave) * 64;
    int b  = Mbase / HoWo;             // constant across the wave
    int cb = b * Cin * HiWi;
    int statbase = b * Cin;

    // Per-M-fragment input-window origin for this lane's matrix row.
    int iy0[4], ix0[4];
#pragma unroll
    for (int f = 0; f < 4; ++f) {
        int m  = Mbase + f * 16 + lo;
        int rr = m - b * HoWo;
        int oy = rr / Wo;
        int ox = rr - oy * Wo;
        iy0[f] = oy * stride - pad;
        ix0[f] = ox * stride - pad;
    }

    v8f zero;
#pragma unroll
    for (int j = 0; j < 8; ++j) zero[j] = 0.0f;
    v8f acc[4][2];
#pragma unroll
    for (int f = 0; f < 4; ++f) { acc[f][0] = zero; acc[f][1] = zero; }

    int kk = hi << 1;                  // this lane holds K slots kk, kk+1
    for (int k0 = 0; k0 < Kpad; k0 += 4) {
        int kA = k0 + kk;
        int kB = kA + 1;
        int pA = ldsT[kA];
        int pB = ldsT[kB];
        int cA = pA >> 16, kyA = (pA >> 8) & 255, kxA = pA & 255;
        int cB = pB >> 16, kyB = (pB >> 8) & 255, kxB = pB & 255;

        v2f bf0, bf1;                  // B fragments for the two N-tiles
        bf0.x = ldsW[(kA << 5) + lo];
        bf0.y = ldsW[(kB << 5) + lo];
        bf1.x = ldsW[(kA << 5) + 16 + lo];
        bf1.y = ldsW[(kB << 5) + 16 + lo];

#pragma unroll
        for (int f = 0; f < 4; ++f) {
            v2f af;
            af.x = load_a<NORM, RELU>(in, nmean, ninv, cb, statbase,
                                      iy0[f], ix0[f], cA, kyA, kxA,
                                      Cin, Hi, Wi, HiWi);
            af.y = load_a<NORM, RELU>(in, nmean, ninv, cb, statbase,
                                      iy0[f], ix0[f], cB, kyB, kxB,
                                      Cin, Hi, Wi, HiWi);
            acc[f][0] = __builtin_amdgcn_wmma_f32_16x16x4_f32(
                false, af, false, bf0, (short)0, acc[f][0], false, false);
            acc[f][1] = __builtin_amdgcn_wmma_f32_16x16x4_f32(
                false, af, false, bf1, (short)0, acc[f][1], false, false);
        }
    }

    // D store: lane (hi,lo), VGPR r holds D[M = frag*16 + hi*8 + r][N = lo].
    float bv0 = bias[lo];
    float bv1 = bias[16 + lo];
    int outb = b * 32 * HoWo;
#pragma unroll
    for (int f = 0; f < 4; ++f) {
        int mb = Mbase + f * 16 + (hi << 3);
        int rr = mb - b * HoWo;
        int oy = rr / Wo;
        int ox = rr - oy * Wo;
#pragma unroll
        for (int r = 0; r < 8; ++r) {
            int base = outb + oy * Wo + ox;
            int i0 = base + lo * HoWo;
            int i1 = base + (16 + lo) * HoWo;
            float v0 = acc[f][0][r] + bv0;
            float v1 = acc[f][1][r] + bv1;
            if (RESIN) { v0 += out[i0]; v1 += out[i1]; }
            out[i0] = v0;
            out[i1] = v1;
            if (++ox == Wo) { ox = 0; ++oy; }   // Ho*Wo % 64 == 0: never leaves b
        }
    }
}

// ---------------------------------------------------------------------------
// Gate logits: a[b,h,o] = sigmoid( sum_t text[b,t]*attn_w[h,o,t] + attn_b[h,o] )
// ---------------------------------------------------------------------------
__global__ __launch_bounds__(256) void gates_kernel(
    const float* __restrict__ text, const float* __restrict__ aw,
    const float* __restrict__ ab, float* __restrict__ g)
{
    int gid = blockIdx.x * 256 + threadIdx.x;      // 64*1280 exactly
    int b  = gid / 1280;
    int ho = gid - b * 1280;
    const float* t = text + b * 256;
    const float* wr = aw + ho * 256;
    float s = ab[ho];
    for (int i = 0; i < 256; ++i) s = fmaf(t[i], wr[i], s);
    g[gid] = 1.0f / (1.0f + expf(-s));
}

// ---------------------------------------------------------------------------
// Per-batch attention tail: scores -> softmax(channels) -> attention maps ->
// final linear head. Everything staged in LDS (f4[b]=32KB, gates=5KB, ...).
// ---------------------------------------------------------------------------
__global__ __launch_bounds__(256) void final_kernel(
    const float* __restrict__ f4, const float* __restrict__ g,
    const float* __restrict__ fw, const float* __restrict__ fb,
    float* __restrict__ out)
{
    __shared__ float sf[8192];   // feat[b]  : 32 channels x 256 spatial
    __shared__ float sg[1280];   // gates[b] : 5 heads x 256 spatial
    __shared__ float sp[160];    // scores / probs : 5 x 32
    __shared__ float sm[1280];   // attention maps : 5 x 256
    int b = blockIdx.x, tid = threadIdx.x;

    for (int i = tid; i < 8192; i += 256) sf[i] = f4[b * 8192 + i];
    for (int i = tid; i < 1280; i += 256) sg[i] = g[b * 1280 + i];
    __syncthreads();

    for (int p = tid; p < 160; p += 256) {
        int h = p >> 5, c = p & 31;
        const float* gh = sg + h * 256;
        const float* fc = sf + c * 256;
        float s = 0.0f;
        for (int i = 0; i < 256; ++i) s = fmaf(gh[i], fc[i], s);
        sp[p] = s;
    }
    __syncthreads();

    if (tid < 5) {
        float mx = -1e30f;
        for (int c = 0; c < 32; ++c) mx = fmaxf(mx, sp[tid * 32 + c]);
        float e[32];
        float ssum = 0.0f;
        for (int c = 0; c < 32; ++c) { e[c] = expf(sp[tid * 32 + c] - mx); ssum += e[c]; }
        float r = 1.0f / ssum;
        for (int c = 0; c < 32; ++c) sp[tid * 32 + c] = e[c] * r;
    }
    __syncthreads();

    for (int q = tid; q < 1280; q += 256) {
        int h = q >> 8, ij = q & 255;
        float s = 0.0f;
        for (int c = 0; c < 32; ++c) s = fmaf(sp[h * 32 + c], sf[c * 256 + ij], s);
        sm[q] = s;
    }
    __syncthreads();

    for (int e = tid; e < 512; e += 256) {
        const float* wr = fw + e * 1280;
        float s = fb[e];
        for (int j = 0; j < 1280; ++j) s = fmaf(sm[j], wr[j], s);
        out[b * 512 + e] = s;
    }
}

// ---------------------------------------------------------------------------
// Host-side orchestration.
// ---------------------------------------------------------------------------
extern "C" void kernel_launch(void* const* d_in, const int* in_sizes, int n_in,
                              void* d_out, int out_size, void* d_ws, size_t ws_size,
                              hipStream_t stream) {
    (void)in_sizes; (void)n_in; (void)out_size; (void)ws_size;
    const float* x      = (const float*)d_in[0];
    const float* text   = (const float*)d_in[1];
    const float* c1w    = (const float*)d_in[2];
    const float* c1b    = (const float*)d_in[3];
    const float* rbs_w1 = (const float*)d_in[4];
    const float* rbs_b1 = (const float*)d_in[5];
    const float* rbs_w2 = (const float*)d_in[6];
    const float* rbs_b2 = (const float*)d_in[7];
    const float* rbs_ws = (const float*)d_in[8];
    const float* rbs_bs = (const float*)d_in[9];
    const float* rb_w1  = (const float*)d_in[10];
    const float* rb_b1  = (const float*)d_in[11];
    const float* rb_w2  = (const float*)d_in[12];
    const float* rb_b2  = (const float*)d_in[13];
    const float* aw     = (const float*)d_in[14];
    const float* ab     = (const float*)d_in[15];
    const float* fw     = (const float*)d_in[16];
    const float* fb     = (const float*)d_in[17];
    float* out = (float*)d_out;

    // Workspace layout (floats): X big feature, Z conv scratch, W3 feature,
    // Sm/Si instance-norm stats, G attention gates. ~202 MB total.
    float* ws = (float*)d_ws;
    float* X  = ws;                 // 33,554,432 floats (64x32x128x128)
    float* Z  = X  + 33554432;      //  8,388,608 floats
    float* W3 = Z  + 8388608;       //  8,388,608 floats
    float* Sm = W3 + 8388608;       //      2,048 floats
    float* Si = Sm + 2048;          //      2,048 floats
    float* G  = Si + 2048;          //     81,920 floats

    dim3 blk(256);

    // ---- stem: feat = conv7x7_s2(IN(x)) --------------------------------
    stats_kernel<<<192, blk, 0, stream>>>(x, Sm, Si, 65536);
    conv_wmma_kernel<true, false, false><<<2048, blk, 0, stream>>>(
        x, c1w, c1b, Sm, Si, X, 3, 7, 7, 2, 3, 256, 256, 128, 128, 147, 148);

    // ---- strided resblock 0: X(128^2) -> W3(64^2), h in Z --------------
    stats_kernel<<<2048, blk, 0, stream>>>(X, Sm, Si, 16384);
    conv_wmma_kernel<true, true, false><<<512, blk, 0, stream>>>(
        X, rbs_w1 + 0 * 9216, rbs_b1 + 0 * 32, Sm, Si, Z,
        32, 3, 3, 2, 1, 128, 128, 64, 64, 288, 288);
    stats_kernel<<<2048, blk, 0, stream>>>(Z, Sm, Si, 4096);
    conv_wmma_kernel<false, false, false><<<512, blk, 0, stream>>>(
        X, rbs_ws + 0 * 1024, rbs_bs + 0 * 32, nullptr, nullptr, W3,
        32, 1, 1, 2, 0, 128, 128, 64, 64, 32, 32);
    conv_wmma_kernel<true, true, true><<<512, blk, 0, stream>>>(
        Z, rbs_w2 + 0 * 9216, rbs_b2 + 0 * 32, Sm, Si, W3,
        32, 3, 3, 1, 1, 64, 64, 64, 64, 288, 288);

    // ---- plain resblock 0 on W3 (64^2) ---------------------------------
    stats_kernel<<<2048, blk, 0, stream>>>(W3, Sm, Si, 4096);
    conv_wmma_kernel<true, true, false><<<512, blk, 0, stream>>>(
        W3, rb_w1 + 0 * 9216, rb_b1 + 0 * 32, Sm, Si, Z,
        32, 3, 3, 1, 1, 64, 64, 64, 64, 288, 288);
    stats_kernel<<<2048, blk, 0, stream>>>(Z, Sm, Si, 4096);
    conv_wmma_kernel<true, true, true><<<512, blk, 0, stream>>>(
        Z, rb_w2 + 0 * 9216, rb_b2 + 0 * 32, Sm, Si, W3,
        32, 3, 3, 1, 1, 64, 64, 64, 64, 288, 288);

    // ---- strided resblock 1: W3(64^2) -> X(32^2), h in Z ---------------
    stats_kernel<<<2048, blk, 0, stream>>>(W3, Sm, Si, 4096);
    conv_wmma_kernel<true, true, false><<<128, blk, 0, stream>>>(
        W3, rbs_w1 + 1 * 9216, rbs_b1 + 1 * 32, Sm, Si, Z,
        32, 3, 3, 2, 1, 64, 64, 32, 32, 288, 288);
    stats_kernel<<<2048, blk, 0, stream>>>(Z, Sm, Si, 1024);
    conv_wmma_kernel<false, false, false><<<128, blk, 0, stream>>>(
        W3, rbs_ws + 1 * 1024, rbs_bs + 1 * 32, nullptr, nullptr, X,
        32, 1, 1, 2, 0, 64, 64, 32, 32, 32, 32);
    conv_wmma_kernel<true, true, true><<<128, blk, 0, stream>>>(
        Z, rbs_w2 + 1 * 9216, rbs_b2 + 1 * 32, Sm, Si, X,
        32, 3, 3, 1, 1, 32, 32, 32, 32, 288, 288);

    // ---- plain resblock 1 on X (32^2) ----------------------------------
    stats_kernel<<<2048, blk, 0, stream>>>(X, Sm, Si, 1024);
    conv_wmma_kernel<true, true, false><<<128, blk, 0, stream>>>(
        X, rb_w1 + 1 * 9216, rb_b1 + 1 * 32, Sm, Si, Z,
        32, 3, 3, 1, 1, 32, 32, 32, 32, 288, 288);
    stats_kernel<<<2048, blk, 0, stream>>>(Z, Sm, Si, 1024);
    conv_wmma_kernel<true, true, true><<<128, blk, 0, stream>>>(
        Z, rb_w2 + 1 * 9216, rb_b2 + 1 * 32, Sm, Si, X,
        32, 3, 3, 1, 1, 32, 32, 32, 32, 288, 288);

    // ---- strided resblock 2: X(32^2) -> W3(16^2), h in Z ---------------
    stats_kernel<<<2048, blk, 0, stream>>>(X, Sm, Si, 1024);
    conv_wmma_kernel<true, true, false><<<32, blk, 0, stream>>>(
        X, rbs_w1 + 2 * 9216, rbs_b1 + 2 * 32, Sm, Si, Z,
        32, 3, 3, 2, 1, 32, 32, 16, 16, 288, 288);
    stats_kernel<<<2048, blk, 0, stream>>>(Z, Sm, Si, 256);
    conv_wmma_kernel<false, false, false><<<32, blk, 0, stream>>>(
        X, rbs_ws + 2 * 1024, rbs_bs + 2 * 32, nullptr, nullptr, W3,
        32, 1, 1, 2, 0, 32, 32, 16, 16, 32, 32);
    conv_wmma_kernel<true, true, true><<<32, blk, 0, stream>>>(
        Z, rbs_w2 + 2 * 9216, rbs_b2 + 2 * 32, Sm, Si, W3,
        32, 3, 3, 1, 1, 16, 16, 16, 16, 288, 288);

    // ---- attention + head ----------------------------------------------
    gates_kernel<<<320, blk, 0, stream>>>(text, aw, ab, G);
    final_kernel<<<64, blk, 0, stream>>>(W3, G, fw, fb, out);
}